// TripletSemiHardLoss_2061584302385
// MI455X (gfx1250) — compile-verified
//
#include <hip/hip_runtime.h>

// TripletSemiHardLoss for MI455X (gfx1250, wave32).
// B=512 embeddings, D=128, fp32. Gram matrix via V_WMMA_F32_16X16X4_F32.

#define BN 512
#define DD 128
#define MARGIN 0.1f

typedef __attribute__((ext_vector_type(2))) float v2f;
typedef __attribute__((ext_vector_type(8))) float v8f;

// workspace layout (in floats)
#define WS_PDIST 0
#define WS_SQ    (BN * BN)
#define WS_NEGIN (WS_SQ + BN)
#define WS_ACC   (WS_NEGIN + BN)   // [0] = loss sum, [1] = num positives

// ---------------------------------------------------------------------------
// Kernel 0: per-row squared norms + zero the accumulators.
// ---------------------------------------------------------------------------
__global__ void sqnorm_init_kernel(const float* __restrict__ E,
                                   float* __restrict__ sq,
                                   float* __restrict__ acc) {
  int i = blockIdx.x * blockDim.x + threadIdx.x;
  if (i < 2) acc[i] = 0.0f;
  if (i < BN) {
    const float* r = E + i * DD;
    float s = 0.0f;
#pragma unroll 8
    for (int k = 0; k < DD; ++k) s += r[k] * r[k];
    sq[i] = s;
  }
}

// ---------------------------------------------------------------------------
// Kernel 1: pdist tiles via fp32 WMMA. One wave per 16x16 output tile.
// G tile = A(16x128) x B(128x16) accumulated over 32 x (K=4) WMMA issues.
// A layout (16x4 f32): lanes 0-15 -> M=lane, VGPR0=K0,VGPR1=K1;
//                      lanes 16-31 -> M=lane-16, VGPR0=K2,VGPR1=K3.
// B layout mirrors A with N in place of M (B = E^T column tile).
// C/D layout: N = lane%16, M = v + 8*(lane>=16).
// ---------------------------------------------------------------------------
__global__ void pdist_wmma_kernel(const float* __restrict__ E,
                                  const float* __restrict__ sq,
                                  float* __restrict__ pd) {
  const int wave = threadIdx.x >> 5;
  const int lane = threadIdx.x & 31;
  const int tile = blockIdx.x * (blockDim.x >> 5) + wave;  // 1024 tiles
  const int ti = tile >> 5;   // row block 0..31
  const int tj = tile & 31;   // col block 0..31
  const int half = lane >> 4; // K sub-group
  const int m = lane & 15;

  const float* arow = E + (ti * 16 + m) * DD + half * 2;
  const float* brow = E + (tj * 16 + m) * DD + half * 2;

  v8f c = {};
#pragma unroll
  for (int k0 = 0; k0 < DD; k0 += 4) {
    v2f a, b;
    a.x = arow[k0];
    a.y = arow[k0 + 1];
    b.x = brow[k0];
    b.y = brow[k0 + 1];
    // D = A x B + C   (fp32, 16x16 tile striped across the wave)
    c = __builtin_amdgcn_wmma_f32_16x16x4_f32(
        /*neg_a=*/false, a, /*neg_b=*/false, b,
        /*c_mod=*/(short)0, c, /*reuse_a=*/false, /*reuse_b=*/false);
  }

  const int col = tj * 16 + m;
  const float sqc = sq[col];
#pragma unroll
  for (int v = 0; v < 8; ++v) {
    const int row = ti * 16 + v + half * 8;
    float d = sq[row] + sqc - 2.0f * c[v];
    d = fmaxf(d, 0.0f);          // clamp + (d>0) mask (idempotent for d>=0)
    if (row == col) d = 0.0f;    // zero diagonal
    pd[row * BN + col] = d;
  }
}

// ---------------------------------------------------------------------------
// Kernel 2: per-anchor fallback negative.
// negatives_inside[j] = max_k((pd[j,k]-row_min)*[label k != label j]) + row_min
//                     = max(0, max_{neg k} pd[j,k] - row_min) + row_min
// One wave per row j; lane-strided scan + shfl_xor reductions (wave32).
// ---------------------------------------------------------------------------
__global__ void row_stats_kernel(const float* __restrict__ pd,
                                 const int* __restrict__ labels,
                                 float* __restrict__ negin) {
  const int wave = threadIdx.x >> 5;
  const int lane = threadIdx.x & 31;
  const int j = blockIdx.x * (blockDim.x >> 5) + wave;
  const int lj = labels[j];
  const float* row = pd + j * BN;

  float mn = 1e30f;      // row min over all k
  float ngmax = -1e30f;  // max over negatives
  for (int k = lane; k < BN; k += 32) {
    const float v = row[k];
    mn = fminf(mn, v);
    if (labels[k] != lj) ngmax = fmaxf(ngmax, v);
  }
#pragma unroll
  for (int off = 16; off > 0; off >>= 1) {
    mn = fminf(mn, __shfl_xor(mn, off, 32));
    ngmax = fmaxf(ngmax, __shfl_xor(ngmax, off, 32));
  }
  if (lane == 0) negin[j] = fmaxf(ngmax - mn, 0.0f) + mn;
}

// ---------------------------------------------------------------------------
// Kernel 3: semi-hard mining + loss accumulation.
// For anchor j and positive i: semi = min{pd[j,k] : label[k]!=lj, pd[j,k]>pd[j,i]}
// (exists) else negatives_inside[j]. loss += max(MARGIN + pd[j,i] - semi, 0).
// One block per anchor; pdist row + labels cached in LDS; waves split i,
// lanes split k with a wave-min reduction.
// ---------------------------------------------------------------------------
__global__ void mine_loss_kernel(const float* __restrict__ pd,
                                 const int* __restrict__ labels,
                                 const float* __restrict__ negin,
                                 float* __restrict__ acc) {
  __shared__ float srow[BN];
  __shared__ int slab[BN];
  const int j = blockIdx.x;
  for (int t = threadIdx.x; t < BN; t += blockDim.x) {
    srow[t] = pd[j * BN + t];
    slab[t] = labels[t];
  }
  __syncthreads();

  const int lj = slab[j];
  const float nin = negin[j];
  const int wave = threadIdx.x >> 5;
  const int lane = threadIdx.x & 31;
  const int nwaves = blockDim.x >> 5;

  float lsum = 0.0f;
  float lcnt = 0.0f;
  for (int i = wave; i < BN; i += nwaves) {       // uniform across the wave
    if (i == j || slab[i] != lj) continue;        // positives only
    const float t = srow[i];
    float mnv = 1e30f;
    for (int k = lane; k < BN; k += 32) {
      const float v = srow[k];
      if (slab[k] != lj && v > t) mnv = fminf(mnv, v);
    }
#pragma unroll
    for (int off = 16; off > 0; off >>= 1)
      mnv = fminf(mnv, __shfl_xor(mnv, off, 32));
    if (lane == 0) {
      const float semi = (mnv < 1e29f) ? mnv : nin;  // mask_final select
      lsum += fmaxf(MARGIN + t - semi, 0.0f);
      lcnt += 1.0f;
    }
  }
  if (lane == 0 && lcnt != 0.0f) {
    atomicAdd(&acc[0], lsum);
    atomicAdd(&acc[1], lcnt);
  }
}

// ---------------------------------------------------------------------------
// Kernel 4: finalize scalar.
// ---------------------------------------------------------------------------
__global__ void finalize_kernel(const float* __restrict__ acc,
                                float* __restrict__ out) {
  if (threadIdx.x == 0 && blockIdx.x == 0) out[0] = acc[0] / acc[1];
}

extern "C" void kernel_launch(void* const* d_in, const int* in_sizes, int n_in,
                              void* d_out, int out_size, void* d_ws, size_t ws_size,
                              hipStream_t stream) {
  (void)in_sizes; (void)n_in; (void)out_size; (void)ws_size;
  const float* emb = (const float*)d_in[0];
  const int* labels = (const int*)d_in[1];
  float* ws = (float*)d_ws;
  float* pd = ws + WS_PDIST;
  float* sq = ws + WS_SQ;
  float* negin = ws + WS_NEGIN;
  float* acc = ws + WS_ACC;
  float* out = (float*)d_out;

  sqnorm_init_kernel<<<(BN + 255) / 256, 256, 0, stream>>>(emb, sq, acc);
  // 32x32 = 1024 tiles of 16x16; 8 waves (tiles) per 256-thread block.
  pdist_wmma_kernel<<<1024 / 8, 256, 0, stream>>>(emb, sq, pd);
  // 8 rows per block (one per wave).
  row_stats_kernel<<<BN / 8, 256, 0, stream>>>(pd, labels, negin);
  mine_loss_kernel<<<BN, 256, 0, stream>>>(pd, labels, negin, acc);
  finalize_kernel<<<1, 32, 0, stream>>>(acc, out);
}